// AdaptiveComputationTime_38070590112081
// MI455X (gfx1250) — compile-verified
//
#include <hip/hip_runtime.h>

// Adaptive Computation Time, collapsed form.
// x is constant across ACT steps => p is constant per row, the state update
// weight is uniform across features, so state = c * x for a per-row scalar c,
// and ponder = n_updates + remainder. One GEMV + scalar scan + scaled copy.
//
// Bandwidth-bound: ~268 MB moved @ 23.3 TB/s => ~11.5 us floor. GEMV mapped
// onto V_WMMA_F32_16X16X4_F32 (fp32 matrix pipe, W in B-matrix column 0).

typedef float v2f __attribute__((ext_vector_type(2)));
typedef float v4f __attribute__((ext_vector_type(4)));
typedef float v8f __attribute__((ext_vector_type(8)));

#define ACT_D 1024
#define ROWS_PER_WAVE 16
#define WAVES_PER_BLOCK 8
#define ROWS_PER_BLOCK (ROWS_PER_WAVE * WAVES_PER_BLOCK)  // 128
#define MAX_STEPS 20
#define THRESHOLD 0.99f

__global__ __launch_bounds__(256) void act_kernel(
    const float* __restrict__ x, const float* __restrict__ W,
    const float* __restrict__ bbias, float* __restrict__ out_state,
    float* __restrict__ out_ponder) {
  __shared__ float wlds[ACT_D];
  __shared__ float coef[ROWS_PER_BLOCK];

  const int tid  = threadIdx.x;
  const int wave = tid >> 5;   // 0..7
  const int lane = tid & 31;   // wave32
  const int rowBase = blockIdx.x * ROWS_PER_BLOCK + wave * ROWS_PER_WAVE;

  // Stage W into LDS (coalesced, 4 floats per thread).
  for (int i = tid; i < ACT_D; i += 256) wlds[i] = W[i];
  __syncthreads();

  // ---------------- Phase 1: dot[row] = x[row,:] . W via f32 WMMA ----------
  // A (16x4 f32): lane m=lane&15 holds row m; VGPR0/1 hold K = khalf, khalf+1
  // where khalf = 2*(lane>>4). B (4x16): VGPR0/1 hold rows K=khalf, khalf+1
  // for N = lane&15; only column N==0 carries W, others zero.
  v8f acc = {};
  const int m     = lane & 15;
  const int khalf = (lane >> 4) << 1;  // 0 or 2
  const float* xrow = x + (size_t)(rowBase + m) * ACT_D;

#pragma unroll 8
  for (int k = 0; k < ACT_D; k += 4) {
    v2f a = *(const v2f*)(xrow + k + khalf);
    float w0 = wlds[k + khalf];
    float w1 = wlds[k + khalf + 1];
    v2f bf;
    bf.x = (m == 0) ? w0 : 0.0f;  // cndmask keeps EXEC all-1s at the WMMA
    bf.y = (m == 0) ? w1 : 0.0f;
    acc = __builtin_amdgcn_wmma_f32_16x16x4_f32(
        /*neg_a=*/false, a, /*neg_b=*/false, bf,
        /*c_mod=*/(short)0, acc, /*reuse_a=*/false, /*reuse_b=*/false);
  }

  // Column N=0 of D lives in lanes 0 and 16: VGPR i -> M=i (lane 0), M=8+i
  // (lane 16). Those two lanes run the exact 20-step scalar ACT scan.
  if (m == 0) {
    const float bias  = bbias[0];
    const int   mbase = (lane >> 4) * 8;  // 0 or 8
#pragma unroll
    for (int i = 0; i < 8; ++i) {
      float dot = acc[i] + bias;
      float p   = 1.0f / (1.0f + expf(-dot));  // sigmoid, fp32

      float hp = 0.0f, rem = 0.0f, nup = 0.0f, cf = 0.0f;
#pragma unroll
      for (int s = 0; s < MAX_STEPS; ++s) {
        float still = (hp < 1.0f) ? 1.0f : 0.0f;
        float nh    = (((hp + p * still) > THRESHOLD) ? 1.0f : 0.0f) * still;
        hp  = hp + p * still;
        rem = rem + nh * (1.0f - hp);
        hp  = hp + nh * rem;
        float uw = p * still + nh * rem;
        cf  = (1.0f - uw) * cf + uw;   // state = cf * x
        nup = nup + still;
      }
      coef[wave * ROWS_PER_WAVE + mbase + i]   = cf;
      out_ponder[(size_t)rowBase + mbase + i]  = nup + rem;
    }
  }
  __syncthreads();

  // ---------------- Phase 2: out = c[row] * x[row,:], coalesced float4 -----
  // Rows were just read by this wave in phase 1 -> L0/L2 hits, no extra HBM.
  for (int r = 0; r < ROWS_PER_WAVE; ++r) {
    const float cf = coef[wave * ROWS_PER_WAVE + r];
    const v4f* src = (const v4f*)(x + (size_t)(rowBase + r) * ACT_D);
    v4f*       dst = (v4f*)(out_state + (size_t)(rowBase + r) * ACT_D);
#pragma unroll
    for (int i = lane; i < ACT_D / 4; i += 32) {
      v4f v = src[i];
      dst[i] = v * cf;
    }
  }
}

extern "C" void kernel_launch(void* const* d_in, const int* in_sizes, int n_in,
                              void* d_out, int out_size, void* d_ws, size_t ws_size,
                              hipStream_t stream) {
  const float* x = (const float*)d_in[0];   // (B,S,D) fp32
  const float* W = (const float*)d_in[1];   // (D,1)   fp32
  const float* b = (const float*)d_in[2];   // (1,)    fp32

  const int  D    = in_sizes[1];                 // 1024
  const long rows = (long)in_sizes[0] / D;       // 32768

  float* out_state  = (float*)d_out;             // rows*D floats
  float* out_ponder = out_state + (size_t)rows * D;  // rows floats

  dim3 block(256);
  dim3 grid((unsigned)(rows / ROWS_PER_BLOCK));  // 256 blocks
  act_kernel<<<grid, block, 0, stream>>>(x, W, b, out_state, out_ponder);
}